// WaveConv2d_23545010717196
// MI455X (gfx1250) — compile-verified
//
#include <hip/hip_runtime.h>
#include <hip/hip_bf16.h>

// sym8-style filters. Note: h0(analysis) = DEC_LO[::-1] == REC_LO, and
// h1(analysis) = DEC_HI[::-1] == REC_HI, so one pair of constant arrays
// serves both the analysis correlation taps and the synthesis filters.
__constant__ float c_f0[8] = {
     0.23037781330885523f,  0.7148465705525415f,   0.6308807679295904f,
    -0.02798376941698385f, -0.18703481171888114f,  0.030841381835986965f,
     0.032883011666982945f, -0.010597401784997278f };
__constant__ float c_f1[8] = {
    -0.010597401784997278f, -0.032883011666982945f, 0.030841381835986965f,
     0.18703481171888114f,  -0.02798376941698385f,  -0.6308807679295904f,
     0.7148465705525415f,   -0.23037781330885523f };

// ---------------------------------------------------------------------------
// Analysis filter bank along one axis (stride 2, pywt 'symmetric' padding).
// One thread produces one (lo, hi) output pair; 8 taps each.
// inStepF: element stride along the filtered axis; inStepO: along the other.
// ---------------------------------------------------------------------------
__global__ void afb1d_k(const float* __restrict__ in, long inImg,
                        int inStepF, int inStepO,
                        int Nf, int No, int Wout,
                        float* __restrict__ out0, long out0Img,
                        float* __restrict__ out1, long out1Img,
                        int outStepF, int outStepO, int nImg)
{
    long idx = (long)blockIdx.x * blockDim.x + threadIdx.x;
    long total = (long)nImg * No * Wout;
    if (idx >= total) return;
    int  j   = (int)(idx % Wout);
    long t   = idx / Wout;
    int  o   = (int)(t % No);
    int  img = (int)(t / No);

    const int pl = (2 * (Wout - 1) - Nf + 8) >> 1;   // left pad (<= 6)
    const int N2 = 2 * Nf;
    const float* src = in + (long)img * inImg + (long)o * inStepO;

    float a0 = 0.f, a1 = 0.f;
#pragma unroll
    for (int k = 0; k < 8; ++k) {
        int ti = 2 * j + k - pl + N2;     // always >= 0
        int m  = ti % N2;                  // symmetric reflection, period 2N
        if (m >= Nf) m = N2 - 1 - m;
        float v = src[(long)m * inStepF];
        a0 = fmaf(c_f0[k], v, a0);
        a1 = fmaf(c_f1[k], v, a1);
    }
    long od = (long)o * outStepO + (long)j * outStepF;
    out0[(long)img * out0Img + od] = a0;
    out1[(long)img * out1Img + od] = a1;
}

// ---------------------------------------------------------------------------
// Synthesis filter bank along one axis (conv_transpose stride 2, pad L-2).
// out[i] = sum over k with (i+6-k) even, n=(i+6-k)/2 in [0,Nin):
//          REC_LO[k]*lo[n] + REC_HI[k]*hi[n]
// Separate strides for lo/hi inputs so ll-trimming is just a sub-window read.
// ---------------------------------------------------------------------------
__global__ void sfb1d_k(const float* __restrict__ lo, long loImg,
                        int loStepF, int loStepO,
                        const float* __restrict__ hi, long hiImg,
                        int hiStepF, int hiStepO,
                        int Nin, int No, int Nout,
                        float* __restrict__ out, long outImg,
                        int outStepF, int outStepO, int nImg)
{
    long idx = (long)blockIdx.x * blockDim.x + threadIdx.x;
    long total = (long)nImg * No * Nout;
    if (idx >= total) return;
    int  i   = (int)(idx % Nout);
    long t   = idx / Nout;
    int  o   = (int)(t % No);
    int  img = (int)(t / No);

    const float* plo = lo + (long)img * loImg + (long)o * loStepO;
    const float* phi = hi + (long)img * hiImg + (long)o * hiStepO;

    float acc = 0.f;
#pragma unroll
    for (int k = 0; k < 8; ++k) {
        int q = i + 6 - k;
        if (q & 1) continue;
        int n = q >> 1;
        if ((unsigned)n >= (unsigned)Nin) continue;
        acc = fmaf(c_f0[k], plo[(long)n * loStepF], acc);
        acc = fmaf(c_f1[k], phi[(long)n * hiStepF], acc);
    }
    out[(long)img * outImg + (long)o * outStepO + (long)i * outStepF] = acc;
}

// ---------------------------------------------------------------------------
// Channel mixing einsum 'bixy,ioxy->boxy' at P spatial positions using
// V_WMMA_F32_16X16X4_F32:  D(16x16) = A(16x4) x B(4x16) + C, chained over
// K = Cin = 64 in 16 steps.  M = batch (16), N = 16-channel tile of Cout.
// A layout (ISA 7.12.2, 32-bit A 16x4): lanes 0-15 -> M, VGPR v -> K=v,
// lanes 16-31 -> K=v+2.  B symmetric; C/D: VGPR r -> M=r (+8 for lanes>=16),
// lane%16 -> N.
// X: plane (b*64+c) has stride xImg elems, position p within plane.
// W: weights (i*64+o)*P + p.  Y: (b*64+o)*yImg + p.
// ---------------------------------------------------------------------------
typedef float v2f __attribute__((ext_vector_type(2)));
typedef float v8f __attribute__((ext_vector_type(8)));

__global__ void mix_wmma(const float* __restrict__ X, long xImg,
                         const float* __restrict__ W,
                         float* __restrict__ Y, long yImg, int P)
{
    const int lane = threadIdx.x & 31;
    const int wave = threadIdx.x >> 5;
    const int tile = blockIdx.x * (blockDim.x >> 5) + wave;
    const int nTiles = P * 4;
    if (tile >= nTiles) return;          // wave-uniform

    const int p  = tile >> 2;            // spatial position
    const int nt = tile & 3;             // which 16-wide slice of Cout
    const int mn = lane & 15;            // M for A loads, N for B loads
    const int hk = (lane >> 4) * 2;      // K offset (0 or 2) for upper half

    const float* xb = X + (long)mn * 64 * xImg + p;              // A row b=mn
    const float* wb = W + (long)(nt * 16 + mn) * P + p;          // B col o

    v8f acc = {0.f, 0.f, 0.f, 0.f, 0.f, 0.f, 0.f, 0.f};
#pragma unroll
    for (int k0 = 0; k0 < 64; k0 += 4) {
        const int ka = k0 + hk;
        v2f a, b;
        a.x = xb[(long)(ka)     * xImg];
        a.y = xb[(long)(ka + 1) * xImg];
        b.x = wb[(long)(ka)     * 64 * P];
        b.y = wb[(long)(ka + 1) * 64 * P];
        acc = __builtin_amdgcn_wmma_f32_16x16x4_f32(
                  false, a, false, b, (short)0, acc, false, false);
    }
    // store D: element r -> b = r + 8*(lane>=16), channel o = nt*16 + (lane%16)
    const int bb = (lane >> 4) * 8;
    const int oo = nt * 16 + mn;
#pragma unroll
    for (int r = 0; r < 8; ++r) {
        Y[((long)((bb + r) * 64 + oo)) * yImg + p] = acc[r];
    }
}

// ---------------------------------------------------------------------------
extern "C" void kernel_launch(void* const* d_in, const int* in_sizes, int n_in,
                              void* d_out, int out_size, void* d_ws, size_t ws_size,
                              hipStream_t stream)
{
    const float* x  = (const float*)d_in[0];   // (16,64,256,256)
    const float* w1 = (const float*)d_in[1];   // (64,64,22,22)
    const float* w2 = (const float*)d_in[2];   // (64,64,22,22)
    float* out = (float*)d_out;                // (16,64,256,256)

    const int NI = 16 * 64;                    // 1024 (b,c) planes
    // level sizes: 256 -> 131 -> 69 -> 38 -> 22
    float* ws = (float*)d_ws;
    long off = 0;
    auto alloc = [&](long elemsPerImg) { float* p = ws + off; off += elemsPerImg * NI; return p; };
    float* bufA   = alloc(33536);  // 256*131 generic mid buffer
    float* bufB   = alloc(33536);
    float* ll1    = alloc(17161);  // 131*131
    float* h1     = alloc(51483);  // 3*131*131
    float* ll2    = alloc(4761);   // 69*69
    float* h2     = alloc(14283);  // 3*69*69
    float* ll3    = alloc(1444);   // 38*38
    float* h3     = alloc(4332);   // 3*38*38
    float* ll4    = alloc(484);    // 22*22
    float* h4     = alloc(1452);   // 3*22*22
    float* outft  = alloc(484);
    float* band0n = alloc(484);
    float* recon4 = alloc(1444);   // 38*38
    float* recon3 = alloc(4900);   // 70*70
    float* recon2 = alloc(17424);  // 132*132

    const int BLK = 256;
    auto grid = [](long total) { return (unsigned)((total + 255) / 256); };

    // ---------------- Forward DWT ----------------
    // L1: W pass 256x256 -> 256x131, then H pass -> 131x131
    afb1d_k<<<grid((long)NI*256*131), BLK, 0, stream>>>(x, 65536, 1, 256, 256, 256, 131,
        bufA, 33536, bufB, 33536, 1, 131, NI);
    afb1d_k<<<grid((long)NI*131*131), BLK, 0, stream>>>(bufA, 33536, 131, 1, 256, 131, 131,
        ll1, 17161, h1, 51483, 131, 1, NI);
    afb1d_k<<<grid((long)NI*131*131), BLK, 0, stream>>>(bufB, 33536, 131, 1, 256, 131, 131,
        h1 + 17161, 51483, h1 + 2*17161, 51483, 131, 1, NI);
    // L2: 131x131 -> 131x69 -> 69x69
    afb1d_k<<<grid((long)NI*131*69), BLK, 0, stream>>>(ll1, 17161, 1, 131, 131, 131, 69,
        bufA, 9039, bufB, 9039, 1, 69, NI);
    afb1d_k<<<grid((long)NI*69*69), BLK, 0, stream>>>(bufA, 9039, 69, 1, 131, 69, 69,
        ll2, 4761, h2, 14283, 69, 1, NI);
    afb1d_k<<<grid((long)NI*69*69), BLK, 0, stream>>>(bufB, 9039, 69, 1, 131, 69, 69,
        h2 + 4761, 14283, h2 + 2*4761, 14283, 69, 1, NI);
    // L3: 69x69 -> 69x38 -> 38x38
    afb1d_k<<<grid((long)NI*69*38), BLK, 0, stream>>>(ll2, 4761, 1, 69, 69, 69, 38,
        bufA, 2622, bufB, 2622, 1, 38, NI);
    afb1d_k<<<grid((long)NI*38*38), BLK, 0, stream>>>(bufA, 2622, 38, 1, 69, 38, 38,
        ll3, 1444, h3, 4332, 38, 1, NI);
    afb1d_k<<<grid((long)NI*38*38), BLK, 0, stream>>>(bufB, 2622, 38, 1, 69, 38, 38,
        h3 + 1444, 4332, h3 + 2*1444, 4332, 38, 1, NI);
    // L4: 38x38 -> 38x22 -> 22x22
    afb1d_k<<<grid((long)NI*38*22), BLK, 0, stream>>>(ll3, 1444, 1, 38, 38, 38, 22,
        bufA, 836, bufB, 836, 1, 22, NI);
    afb1d_k<<<grid((long)NI*22*22), BLK, 0, stream>>>(bufA, 836, 22, 1, 38, 22, 22,
        ll4, 484, h4, 1452, 22, 1, NI);
    afb1d_k<<<grid((long)NI*22*22), BLK, 0, stream>>>(bufB, 836, 22, 1, 38, 22, 22,
        h4 + 484, 1452, h4 + 2*484, 1452, 22, 1, NI);

    // ---------------- WMMA channel mixing ----------------
    // 484 positions x 4 N-tiles = 1936 waves per call; 8 waves/block -> 242 blocks.
    mix_wmma<<<242, 256, 0, stream>>>(ll4, 484, w1, outft, 484, 484);
    mix_wmma<<<242, 256, 0, stream>>>(h4, 1452, w2, band0n, 484, 484);

    // ---------------- Inverse DWT ----------------
    // L4 inv: (outft, band0n) + (hl,hh of h4) -> 38x38
    sfb1d_k<<<grid((long)NI*22*38), BLK, 0, stream>>>(outft, 484, 22, 1,
        band0n, 484, 22, 1, 22, 22, 38, bufA, 836, 22, 1, NI);
    sfb1d_k<<<grid((long)NI*22*38), BLK, 0, stream>>>(h4 + 484, 1452, 22, 1,
        h4 + 2*484, 1452, 22, 1, 22, 22, 38, bufB, 836, 22, 1, NI);
    sfb1d_k<<<grid((long)NI*38*38), BLK, 0, stream>>>(bufA, 836, 1, 22,
        bufB, 836, 1, 22, 22, 38, 38, recon4, 1444, 1, 38, NI);
    // L3 inv: recon4(38) + h3(38) -> 70x70
    sfb1d_k<<<grid((long)NI*38*70), BLK, 0, stream>>>(recon4, 1444, 38, 1,
        h3, 4332, 38, 1, 38, 38, 70, bufA, 2660, 38, 1, NI);
    sfb1d_k<<<grid((long)NI*38*70), BLK, 0, stream>>>(h3 + 1444, 4332, 38, 1,
        h3 + 2*1444, 4332, 38, 1, 38, 38, 70, bufB, 2660, 38, 1, NI);
    sfb1d_k<<<grid((long)NI*70*70), BLK, 0, stream>>>(bufA, 2660, 1, 38,
        bufB, 2660, 1, 38, 38, 70, 70, recon3, 4900, 1, 70, NI);
    // L2 inv: recon3 trimmed 70->69 + h2(69) -> 132x132
    sfb1d_k<<<grid((long)NI*69*132), BLK, 0, stream>>>(recon3, 4900, 70, 1,
        h2, 14283, 69, 1, 69, 69, 132, bufA, 9108, 69, 1, NI);
    sfb1d_k<<<grid((long)NI*69*132), BLK, 0, stream>>>(h2 + 4761, 14283, 69, 1,
        h2 + 2*4761, 14283, 69, 1, 69, 69, 132, bufB, 9108, 69, 1, NI);
    sfb1d_k<<<grid((long)NI*132*132), BLK, 0, stream>>>(bufA, 9108, 1, 69,
        bufB, 9108, 1, 69, 69, 132, 132, recon2, 17424, 1, 132, NI);
    // L1 inv: recon2 trimmed 132->131 + h1(131) -> 256x256 (final output)
    sfb1d_k<<<grid((long)NI*131*256), BLK, 0, stream>>>(recon2, 17424, 132, 1,
        h1, 51483, 131, 1, 131, 131, 256, bufA, 33536, 131, 1, NI);
    sfb1d_k<<<grid((long)NI*131*256), BLK, 0, stream>>>(h1 + 17161, 51483, 131, 1,
        h1 + 2*17161, 51483, 131, 1, 131, 131, 256, bufB, 33536, 131, 1, NI);
    sfb1d_k<<<grid((long)NI*256*256), BLK, 0, stream>>>(bufA, 33536, 1, 131,
        bufB, 33536, 1, 131, 131, 256, 256, out, 65536, 1, 256, NI);

    (void)in_sizes; (void)n_in; (void)out_size; (void)ws_size;
}